// Model_15599321219084
// MI455X (gfx1250) — compile-verified
//
#include <hip/hip_runtime.h>
#include <stdint.h>

#define NTG    20000
#define NGO    1000
#define NCELLS 512
#define NCUTS  200000
#define NH     128
#define NW     127
#define MINW   0.001f
#define MINH   0.001f
#define WPB    8      // waves per block
#define NBLK_B 1024
#define LOG_NTG 9.9034875525361272f

__device__ __forceinline__ float wave_fmax(float v) {
#pragma unroll
  for (int o = 16; o > 0; o >>= 1) v = fmaxf(v, __shfl_xor(v, o, 32));
  return v;
}
__device__ __forceinline__ float wave_fsum(float v) {
#pragma unroll
  for (int o = 16; o > 0; o >>= 1) v += __shfl_xor(v, o, 32);
  return v;
}
__device__ __forceinline__ int wave_isum(int v) {
#pragma unroll
  for (int o = 16; o > 0; o >>= 1) v += __shfl_xor(v, o, 32);
  return v;
}
// slot is wave-uniform
__device__ __forceinline__ float pick4(float a0, float a1, float a2, float a3, int slot) {
  float r = a0;
  r = (slot == 1) ? a1 : r;
  r = (slot == 2) ? a2 : r;
  r = (slot == 3) ? a3 : r;
  return r;
}
// exp(uh + slope*sigmoid(exp(scale)*lat + shift))
__device__ __forceinline__ float height_exp(float uh, float sl, float sc, float sh, float lat) {
  float t = __expf(sc) * lat + sh;
  float sg = 1.0f / (1.0f + __expf(-t));
  return __expf(uh + sl * sg);
}

// async-stage 4 gene rows (4 x 512B) into LDS at ldsdst
__device__ __forceinline__ void stage_rows_async(
    const float* __restrict__ sp_h, const float* __restrict__ slope_w,
    const float* __restrict__ scale_w, const float* __restrict__ shift_w,
    int g, uint32_t ldsdst, int lane) {
  uint64_t off = (uint64_t)((size_t)g * NH * sizeof(float)) + (uint64_t)(lane * 16);
  uint64_t a0 = (uint64_t)(uintptr_t)sp_h    + off;
  uint64_t a1 = (uint64_t)(uintptr_t)slope_w + off;
  uint64_t a2 = (uint64_t)(uintptr_t)scale_w + off;
  uint64_t a3 = (uint64_t)(uintptr_t)shift_w + off;
  uint32_t l0 = ldsdst + (uint32_t)(lane * 16);
  uint32_t l1 = l0 + 512u, l2 = l0 + 1024u, l3 = l0 + 1536u;
  asm volatile("global_load_async_to_lds_b128 %0, %1, off" :: "v"(l0), "v"(a0) : "memory");
  asm volatile("global_load_async_to_lds_b128 %0, %1, off" :: "v"(l1), "v"(a1) : "memory");
  asm volatile("global_load_async_to_lds_b128 %0, %1, off" :: "v"(l2), "v"(a2) : "memory");
  asm volatile("global_load_async_to_lds_b128 %0, %1, off" :: "v"(l3), "v"(a3) : "memory");
}

// ---------------- Kernel A: per-cell logsumexp over all 20000 genes ----------
__global__ __launch_bounds__(256) void lse_kernel(
    const float* __restrict__ baseline, const float* __restrict__ slopeO,
    const float* __restrict__ latent, float* __restrict__ lse) {
  __shared__ float red[256];
  const int c = blockIdx.x;
  const int t = threadIdx.x;
  const float lat = latent[c];
  float m = -1e30f;
  for (int g = t; g < NTG; g += 256) m = fmaxf(m, baseline[g] + slopeO[g] * lat);
  red[t] = m; __syncthreads();
  for (int s = 128; s > 0; s >>= 1) { if (t < s) red[t] = fmaxf(red[t], red[t + s]); __syncthreads(); }
  m = red[0]; __syncthreads();
  float sum = 0.f;
  for (int g = t; g < NTG; g += 256) sum += __expf(baseline[g] + slopeO[g] * lat - m);
  red[t] = sum; __syncthreads();
  for (int s = 128; s > 0; s >>= 1) { if (t < s) red[t] += red[t + s]; __syncthreads(); }
  if (t == 0) lse[c] = m + __logf(red[0]);
}

// ---------------- Kernel B: one cut per wave, double-buffered async staging --
__global__ __launch_bounds__(256) void cuts_kernel(
    const float* __restrict__ cut_x, const float* __restrict__ latent,
    const float* __restrict__ slope_w, const float* __restrict__ scale_w,
    const float* __restrict__ shift_w, const float* __restrict__ slopeO,
    const float* __restrict__ baseline, const float* __restrict__ sp_h,
    const float* __restrict__ sp_w, const int* __restrict__ genes_oi,
    const int* __restrict__ cxg_ix, const float* __restrict__ lse,
    float* __restrict__ partial) {
  __shared__ __align__(16) float stage[WPB][2][4 * NH];  // double-buffered rows
  __shared__ float wacc[WPB];
  const int lane = threadIdx.x & 31;
  const int wv   = threadIdx.x >> 5;
  const int gwave = blockIdx.x * WPB + wv;
  const int nwaves = gridDim.x * WPB;
  const uint32_t base0 = (uint32_t)(uintptr_t)(&stage[wv][0][0]);
  const uint32_t base1 = (uint32_t)(uintptr_t)(&stage[wv][1][0]);
  const int j0 = lane * 4;

  float acc = 0.f;
  int cut = gwave;
  int buf = 0;
  int   cur_cxg = 0, cur_g = 0;
  float cur_x = 0.f;
  if (cut < NCUTS) {             // prologue: stage first cut into buffer 0
    cur_cxg = cxg_ix[cut];
    cur_x   = cut_x[cut];
    cur_g   = genes_oi[cur_cxg % NGO];
    stage_rows_async(sp_h, slope_w, scale_w, shift_w, cur_g, base0, lane);
  }

  for (; cut < NCUTS; cut += nwaves) {
    const int   c   = cur_cxg / NGO;
    const int   g   = cur_g;
    const float x   = cur_x;
    const float lat = latent[c];

    // --- prefetch next cut: async rows into other buffer + width-row hint ---
    const int nxt = cut + nwaves;
    const bool has_next = (nxt < NCUTS);        // wave-uniform
    int nxt_cxg = 0, nxt_g = 0; float nxt_x = 0.f;
    if (has_next) {
      nxt_cxg = cxg_ix[nxt];
      nxt_x   = cut_x[nxt];
      nxt_g   = genes_oi[nxt_cxg % NGO];
      stage_rows_async(sp_h, slope_w, scale_w, shift_w, nxt_g,
                       buf ? base0 : base1, lane);
      __builtin_prefetch(sp_w + (size_t)nxt_g * NW + j0, 0, 3);  // global_prefetch_b8
    }

    // --- width softmax (127 bins) from global, overlapping async traffic ----
    const float* wrow = sp_w + (size_t)g * NW;
    float uw0 = wrow[j0];
    float uw1 = wrow[j0 + 1];
    float uw2 = wrow[j0 + 2];
    float uw3 = (j0 + 3 < NW) ? wrow[j0 + 3] : -1e30f;   // lane31 pads slot 127
    float wm = wave_fmax(fmaxf(fmaxf(uw0, uw1), fmaxf(uw2, uw3)));
    float e0 = __expf(uw0 - wm), e1 = __expf(uw1 - wm), e2 = __expf(uw2 - wm);
    float e3 = (j0 + 3 < NW) ? __expf(uw3 - wm) : 0.f;
    float inv = 1.0f / wave_fsum(e0 + e1 + e2 + e3);
    const float C1 = 1.0f - MINW * (float)NW;
    float w0 = MINW + C1 * e0 * inv;
    float w1 = MINW + C1 * e1 * inv;
    float w2 = MINW + C1 * e2 * inv;
    float w3 = (j0 + 3 < NW) ? (MINW + C1 * e3 * inv) : 0.f;

    // --- wait: drain current buffer's 4 loads (in-order), keep prefetch -----
    if (has_next) asm volatile("s_wait_asynccnt 0x4" ::: "memory");
    else          asm volatile("s_wait_asynccnt 0x0" ::: "memory");

    const float4* ph = (const float4*)(&stage[wv][buf][0]);
    float4 hh = ph[lane];        // spline_heights row
    float4 sl = ph[32 + lane];   // delta_height_slope row
    float4 sc = ph[64 + lane];   // delta_height_scale row
    float4 sh = ph[96 + lane];   // delta_height_shift row

    // --- h_exp = exp(spline_heights + delta_height) ------------------------
    float h0 = height_exp(hh.x, sl.x, sc.x, sh.x, lat);
    float h1 = height_exp(hh.y, sl.y, sc.y, sh.y, lat);
    float h2 = height_exp(hh.z, sl.z, sc.z, sh.z, lat);
    float h3 = height_exp(hh.w, sl.w, sc.w, sh.w, lat);

    // --- trapezoid area: needs h[j0+4] from next lane ----------------------
    float hnext = __shfl_down(h0, 1, 32);            // lane31: w3==0 masks it
    float a_loc = 0.5f * ((h0 + h1) * w0 + (h1 + h2) * w1 + (h2 + h3) * w2 +
                          (h3 + hnext) * w3);
    float inva = 1.0f / wave_fsum(a_loc);
    const float C2 = 1.0f - MINH;
    float g0 = MINH + C2 * h0 * inva;
    float g1 = MINH + C2 * h1 * inva;
    float g2 = MINH + C2 * h2 * inva;
    float g3 = MINH + C2 * h3 * inva;

    // --- inclusive cumsum of widths (lane-local + shfl_up scan) ------------
    float p0 = w0, p1 = p0 + w1, p2 = p1 + w2, p3 = p2 + w3;
    float incl = p3;
#pragma unroll
    for (int o = 1; o < 32; o <<= 1) {
      float tt = __shfl_up(incl, o, 32);
      if (lane >= o) incl += tt;
    }
    float excl = incl - p3;
    float cs0 = excl + p0, cs1 = excl + p1, cs2 = excl + p2, cs3 = excl + p3;

    // --- bin index: count of j in [0,125] with x >= cumsum[j] --------------
    int cnt = 0;
    cnt += (j0     <= 125 && x >= cs0) ? 1 : 0;
    cnt += (j0 + 1 <= 125 && x >= cs1) ? 1 : 0;
    cnt += (j0 + 2 <= 125 && x >= cs2) ? 1 : 0;
    cnt += (j0 + 3 <= 125 && x >= cs3) ? 1 : 0;
    int idx = wave_isum(cnt);                        // in [0,126], wave-uniform

    // --- gather bin quantities via shfl (uniform index) --------------------
    float w_sel = __shfl(pick4(w0, w1, w2, w3, idx & 3), idx >> 2, 32);
    float h_l   = __shfl(pick4(g0, g1, g2, g3, idx & 3), idx >> 2, 32);
    int ip1 = idx + 1;
    float h_r   = __shfl(pick4(g0, g1, g2, g3, ip1 & 3), ip1 >> 2, 32);
    float loc_l = 0.f;
    if (idx > 0) {
      int im1 = idx - 1;
      loc_l = __shfl(pick4(cs0, cs1, cs2, cs3, im1 & 3), im1 >> 2, 32);
    }

    float alpha = (x - loc_l) / w_sel;
    float hl_log = __logf(alpha * (h_r - h_l) + h_l);
    float overall = baseline[g] + slopeO[g] * lat - lse[c] + LOG_NTG;
    acc += hl_log + overall;

    // --- rotate pipeline ----------------------------------------------------
    cur_cxg = nxt_cxg; cur_x = nxt_x; cur_g = nxt_g;
    buf ^= 1;
  }

  if (lane == 0) wacc[wv] = acc;
  __syncthreads();
  if (threadIdx.x == 0) {
    float s = 0.f;
    for (int i = 0; i < WPB; ++i) s += wacc[i];
    partial[blockIdx.x] = s;
  }
}

// ---------------- Kernel C: deterministic final reduction --------------------
__global__ __launch_bounds__(256) void reduce_kernel(
    const float* __restrict__ partial, int n, float* __restrict__ out) {
  __shared__ float red[256];
  const int t = threadIdx.x;
  float s = 0.f;
  for (int i = t; i < n; i += 256) s += partial[i];
  red[t] = s; __syncthreads();
  for (int k = 128; k > 0; k >>= 1) { if (t < k) red[t] += red[t + k]; __syncthreads(); }
  if (t == 0) out[0] = -red[0];
}

extern "C" void kernel_launch(void* const* d_in, const int* in_sizes, int n_in,
                              void* d_out, int out_size, void* d_ws, size_t ws_size,
                              hipStream_t stream) {
  const float* cut_x    = (const float*)d_in[0];
  const float* latent   = (const float*)d_in[1];
  const float* slope_w  = (const float*)d_in[2];
  const float* scale_w  = (const float*)d_in[3];
  const float* shift_w  = (const float*)d_in[4];
  const float* slopeO   = (const float*)d_in[5];
  const float* baseline = (const float*)d_in[6];
  const float* sp_h     = (const float*)d_in[7];
  const float* sp_w     = (const float*)d_in[8];
  const int*   genes_oi = (const int*)d_in[9];
  const int*   cxg_ix   = (const int*)d_in[10];
  // d_in[11] (cut_localcellxgene_ix) and d_in[12] (cut_local_gene_ix) are
  // redundant given cxg_ix + genes_oi.

  float* ws      = (float*)d_ws;
  float* lse     = ws;             // 512 floats
  float* partial = ws + NCELLS;    // NBLK_B floats

  lse_kernel<<<NCELLS, 256, 0, stream>>>(baseline, slopeO, latent, lse);
  cuts_kernel<<<NBLK_B, 256, 0, stream>>>(cut_x, latent, slope_w, scale_w, shift_w,
                                          slopeO, baseline, sp_h, sp_w,
                                          genes_oi, cxg_ix, lse, partial);
  reduce_kernel<<<1, 256, 0, stream>>>(partial, NBLK_B, (float*)d_out);
}